// HintGenKernelVectorized_8057358647763
// MI455X (gfx1250) — compile-verified
//
#include <hip/hip_runtime.h>

// Segment XOR reduction (parity accumulation) for MI455X / gfx1250.
//
// parities[h][k] = XOR over all i with hint_ids[i]==h of entries[entry_indices[i]][k]
//
// entries:        [NUM_ENTRIES, 5] int64   (d_in[0])
// entry_indices:  [N]             int64    (d_in[1])
// hint_ids:       [N]             int64    (d_in[2])
// num_hints:      scalar          int64    (d_in[3], implied by out_size)
// out:            [num_hints, 5]  int64    (d_out)
//
// Strategy: memory-bound scatter with L2-resident accumulator.
//  - d_out (2.5 MB) zero-initialized, lives in the 192 MB L2.
//  - One thread per index: NT-hinted streaming loads of the index/hint
//    arrays, cached gather of the 40 B entry row, then 5 non-returning
//    GLOBAL_ATOMIC_XOR_B64 into the accumulator.

static constexpr int ROW = 5; // int64 words per entry row

__global__ void __launch_bounds__(256)
zero_u64_kernel(unsigned long long* __restrict__ out, int n) {
    int i = blockIdx.x * blockDim.x + threadIdx.x;
    if (i < n) out[i] = 0ull;
}

__global__ void __launch_bounds__(256)
seg_xor_kernel(const unsigned long long* __restrict__ entries,
               const unsigned long long* __restrict__ entry_indices,
               const unsigned long long* __restrict__ hint_ids,
               unsigned long long* __restrict__ out,
               int n) {
    int i = blockIdx.x * blockDim.x + threadIdx.x;
    if (i >= n) return;

    // Streaming, read-once arrays: non-temporal so they do not evict the
    // L2-resident entries table / accumulator.
    unsigned long long e = __builtin_nontemporal_load(&entry_indices[i]);
    unsigned long long h = __builtin_nontemporal_load(&hint_ids[i]);

    const unsigned long long* __restrict__ row = entries + e * (unsigned long long)ROW;
    unsigned long long* __restrict__ acc = out + h * (unsigned long long)ROW;

    // 40-byte row is only 8-byte aligned -> five b64 loads (compiler may
    // fuse adjacent pairs where legal). Entries table (40 MB) stays hot in
    // the 192 MB L2 across the ~4x reuse per row.
    unsigned long long r0 = row[0];
    unsigned long long r1 = row[1];
    unsigned long long r2 = row[2];
    unsigned long long r3 = row[3];
    unsigned long long r4 = row[4];

    // Non-returning 64-bit XOR atomics: resolved in the L2 atomic units,
    // tracked on STOREcnt, no return data path needed.
    atomicXor(&acc[0], r0);
    atomicXor(&acc[1], r1);
    atomicXor(&acc[2], r2);
    atomicXor(&acc[3], r3);
    atomicXor(&acc[4], r4);
}

extern "C" void kernel_launch(void* const* d_in, const int* in_sizes, int n_in,
                              void* d_out, int out_size, void* d_ws, size_t ws_size,
                              hipStream_t stream) {
    const unsigned long long* entries       = (const unsigned long long*)d_in[0];
    const unsigned long long* entry_indices = (const unsigned long long*)d_in[1];
    const unsigned long long* hint_ids      = (const unsigned long long*)d_in[2];
    // d_in[3] (num_hints) is implied by out_size = num_hints * 5.

    unsigned long long* out = (unsigned long long*)d_out;
    const int n = in_sizes[1]; // N = number of (index, hint) pairs

    // Phase 1: zero the accumulator (harness poisons d_out to 0xAA).
    {
        int blocks = (out_size + 255) / 256;
        zero_u64_kernel<<<blocks, 256, 0, stream>>>(out, out_size);
    }

    // Phase 2: gather + atomic-XOR scatter.
    {
        int blocks = (n + 255) / 256;
        seg_xor_kernel<<<blocks, 256, 0, stream>>>(entries, entry_indices,
                                                   hint_ids, out, n);
    }
}